// LTCCell_33088428048998
// MI455X (gfx1250) — compile-verified
//
#include <hip/hip_runtime.h>
#include <hip/hip_bf16.h>

typedef float v2f __attribute__((ext_vector_type(2)));
typedef float v8f __attribute__((ext_vector_type(8)));

#define BATCH 64
#define SEQ   1024
#define DIN   256
#define UNITS 512

// ---------------- CDNA5 helpers ----------------

__device__ __forceinline__ v8f wmma_f32(v2f a, v2f b, v8f c) {
  // D = A(16x4 f32) * B(4x16 f32) + C(16x16 f32)
  return __builtin_amdgcn_wmma_f32_16x16x4_f32(
      /*neg_a=*/false, a, /*neg_b=*/false, b,
      /*c_mod=*/(short)0, c, /*reuse_a=*/false, /*reuse_b=*/false);
}

// Generic pointer into LDS: low 32 bits are the LDS byte offset (aperture
// occupies the high word only, per CDNA5 aperture mapping).
__device__ __forceinline__ uint32_t lds_off(const void* p) {
  return (uint32_t)(uintptr_t)p;
}

// Async global -> LDS, 16 bytes per lane, tracked with ASYNCcnt.
__device__ __forceinline__ void async_g2l_b128(uint32_t ldsoff, const void* gptr) {
  asm volatile("global_load_async_to_lds_b128 %0, %1, off"
               :: "v"(ldsoff), "v"(gptr)
               : "memory");
}

__device__ __forceinline__ void wait_asynccnt0() {
  asm volatile("s_wait_asynccnt 0" ::: "memory");
}

// ---------------- Kernel 1: xW + b  (big GEMM, memory-bound) ----------------
// x: [BATCH*SEQ, DIN] f32, W: [DIN, UNITS], bias: [UNITS]
// out: [BATCH*SEQ, UNITS]   (d_out, later overwritten in-place by the scan)
// Tiling: 64x64 per workgroup, 8 waves -> each wave one 16x32 tile (2 WMMA
// accumulators), K looped in steps of 4 with v_wmma_f32_16x16x4_f32.

__global__ __launch_bounds__(256)
void ltc_xw_kernel(const float* __restrict__ x, const float* __restrict__ W,
                   const float* __restrict__ bias, float* __restrict__ out) {
  const int lane = threadIdx.x & 31;
  const int wv   = threadIdx.x >> 5;     // 0..7
  const int mt   = wv >> 1;              // 0..3 : 16-row tile within WG
  const int nt   = wv & 1;               // 0..1 : 32-col tile within WG

  const int rowbase = blockIdx.x * 64 + mt * 16;   // 0..65535
  const int colbase = blockIdx.y * 64 + nt * 32;   // 0..511

  const int m   = lane & 15;             // A row
  const int n   = lane & 15;             // B/C col
  const int kk2 = (lane >> 4) * 2;       // K sub-offset {0,2}
  const int mh  = (lane >> 4) * 8;       // C row half offset

  v8f acc0 = {};
  v8f acc1 = {};

  const float* xrow = x + (size_t)(rowbase + m) * DIN;

#pragma unroll 4
  for (int k = 0; k < DIN; k += 4) {
    v2f a = *(const v2f*)(xrow + k + kk2);
    const float* Wk = W + (size_t)(k + kk2) * UNITS + colbase + n;
    v2f b0, b1;
    b0[0] = Wk[0];          b0[1] = Wk[UNITS];          // cols [colbase, +16)
    b1[0] = Wk[16];         b1[1] = Wk[UNITS + 16];     // cols [colbase+16, +32)
    acc0 = wmma_f32(a, b0, acc0);
    acc1 = wmma_f32(a, b1, acc1);
  }

  const float bia0 = bias[colbase + n];
  const float bia1 = bias[colbase + 16 + n];

#pragma unroll
  for (int r = 0; r < 8; ++r) {
    float* o = out + (size_t)(rowbase + mh + r) * UNITS + colbase + n;
    o[0]  = acc0[r] + bia0;
    o[16] = acc1[r] + bia1;
  }
}

// ---------------- Kernel 2: persistent LTC scan ----------------
// 64 workgroups: bt = wg>>4 picks batch tile (16 rows), cs = wg&15 picks a
// 32-column slice of UNITS. U[:,slice] lives in LDS (pad stride 40 floats:
// 16B-aligned rows, bank-conflict-free). Each step:
//   async-copy h[16,512] (ping-pong L2 buffer) -> LDS, split-K WMMA across
//   8 waves, LDS reduce, elementwise LTC update, write h to d_out in place,
//   then sync the 16-WG group with an agent-scope atomic counter.

#define US_STRIDE 40    // floats; 160B rows (16B aligned), conflict-free
#define HS_STRIDE 516   // floats; 2064B rows (16B aligned), conflict-free

__global__ __launch_bounds__(256)
void ltc_scan_kernel(const float* __restrict__ U, const float* __restrict__ tau,
                     float* __restrict__ out, float* __restrict__ hbuf,
                     unsigned* __restrict__ counters) {
  __shared__ float Us[UNITS * US_STRIDE];   // 81920 B : U column slice
  __shared__ float hs[16 * HS_STRIDE];      // 33024 B : staged h tile
  __shared__ v8f   red[8][32];              //  8192 B : split-K partials

  const int wg   = blockIdx.x;
  const int bt   = wg >> 4;        // batch tile: rows [bt*16, bt*16+16)
  const int cs   = wg & 15;        // column slice
  const int c0   = cs * 32;
  const int tid  = threadIdx.x;
  const int lane = tid & 31;
  const int wv   = tid >> 5;       // 0..7
  const int ntl  = wv & 1;         // N-tile (0 or 1)
  const int kq   = wv >> 1;        // K quarter (0..3)

  const int m   = lane & 15;
  const int n   = lane & 15;
  const int kk2 = (lane >> 4) * 2;
  const int mh  = (lane >> 4) * 8;

  const int col = c0 + ntl * 16 + n;         // global unit column
  const float it  = 1.0f / tau[col];         // inv_tau
  const float omt = 1.0f - it;

  // ---- stage U[:, c0:c0+32] into LDS via async tensor path ----
  // 512 rows x 128B; each async op moves 16B/lane => 4 rows per wave-op.
#pragma unroll
  for (int i = 0; i < 16; ++i) {
    const int row = wv * 64 + i * 4 + (lane >> 3);   // 0..511
    const int ch  = lane & 7;                        // 16B chunk in row
    async_g2l_b128(lds_off(&Us[row * US_STRIDE + ch * 4]),
                   U + (size_t)row * UNITS + c0 + ch * 4);
  }
  wait_asynccnt0();
  __syncthreads();

  for (int t = 0; t < SEQ; ++t) {
    const float* rbuf = hbuf + ((size_t)bt * 2 + (t & 1)) * (16 * UNITS);
    float*       wbuf = hbuf + ((size_t)bt * 2 + ((t + 1) & 1)) * (16 * UNITS);

    // prefetch this step's xW slice while h is staged / multiplied
    {
      const size_t idx0 = ((size_t)(bt * 16 + mh) * SEQ + t) * UNITS + col;
      __builtin_prefetch(&out[idx0], 0, 0);
      __builtin_prefetch(&out[idx0 + (size_t)4 * SEQ * UNITS], 0, 0);
    }

    // ---- async-copy h[16,512] -> LDS (32KB, 8 b128 ops per thread) ----
#pragma unroll
    for (int i = 0; i < 8; ++i) {
      const int idx = i * 256 + tid;        // 0..2047 16B-chunks
      const int row = idx >> 7;             // 0..15
      const int ch  = idx & 127;            // 0..127
      async_g2l_b128(lds_off(&hs[row * HS_STRIDE + ch * 4]),
                     rbuf + row * UNITS + ch * 4);
    }
    wait_asynccnt0();
    __syncthreads();

    // ---- h @ U[:,slice] : split-K across waves ----
    v8f acc = {};
    const int kbase = kq * 128;
#pragma unroll 8
    for (int k = 0; k < 128; k += 4) {
      v2f a = *(const v2f*)&hs[m * HS_STRIDE + kbase + k + kk2];
      v2f b;
      b[0] = Us[(kbase + k + kk2) * US_STRIDE + ntl * 16 + n];
      b[1] = Us[(kbase + k + kk2 + 1) * US_STRIDE + ntl * 16 + n];
      acc = wmma_f32(a, b, acc);
    }
    red[wv][lane] = acc;
    __syncthreads();

    if (wv < 2) {
      // reduce 4 K-quarters for this N-tile
      v8f s = red[wv][lane];
      s += red[wv + 2][lane];
      s += red[wv + 4][lane];
      s += red[wv + 6][lane];

      // ---- elementwise LTC update + stores ----
#pragma unroll
      for (int r = 0; r < 8; ++r) {
        const int mrow = mh + r;                         // 0..15 within tile
        const size_t oidx =
            ((size_t)(bt * 16 + mrow) * SEQ + t) * UNITS + col;
        const float xwb  = out[oidx];                    // xW + b (kernel 1)
        const float hold = hs[mrow * HS_STRIDE + col];
        const float hnew = omt * hold + it * (xwb + s[r]);
        out[oidx] = hnew;                                // return_sequences
        wbuf[mrow * UNITS + col] = hnew;                 // carry for peers
      }
    }

    // ---- sync the 16 workgroups of this batch-tile group ----
    __builtin_amdgcn_fence(__ATOMIC_RELEASE, "agent");
    __syncthreads();
    if (tid == 0) {
      __hip_atomic_fetch_add(&counters[bt], 1u,
                             __ATOMIC_RELAXED, __HIP_MEMORY_SCOPE_AGENT);
      const unsigned target = 16u * (unsigned)(t + 1);
      while (__hip_atomic_load(&counters[bt], __ATOMIC_RELAXED,
                               __HIP_MEMORY_SCOPE_AGENT) < target) {
        __builtin_amdgcn_s_sleep(8);
      }
    }
    __syncthreads();
    __builtin_amdgcn_fence(__ATOMIC_ACQUIRE, "agent");
  }
}

// ---------------- host launch ----------------

extern "C" void kernel_launch(void* const* d_in, const int* in_sizes, int n_in,
                              void* d_out, int out_size, void* d_ws, size_t ws_size,
                              hipStream_t stream) {
  const float* x    = (const float*)d_in[0];   // [64,1024,256]
  const float* W    = (const float*)d_in[1];   // [256,512]
  const float* U    = (const float*)d_in[2];   // [512,512]
  const float* b    = (const float*)d_in[3];   // [512]
  const float* tau  = (const float*)d_in[4];   // [512]
  float*       out  = (float*)d_out;           // [64,1024,512]

  // workspace: h ping-pong [4 batch-tiles][2][16][512] f32 + 4 counters
  const size_t HBUF_BYTES = (size_t)4 * 2 * 16 * UNITS * sizeof(float); // 256KB
  float*    hbuf     = (float*)d_ws;
  unsigned* counters = (unsigned*)((char*)d_ws + HBUF_BYTES);

  // zero h0 and the sync counters (capture-legal stream op)
  hipMemsetAsync(d_ws, 0, HBUF_BYTES + 4 * sizeof(unsigned), stream);

  // 1) xW + b into d_out
  dim3 g1((BATCH * SEQ) / 64, UNITS / 64);   // 1024 x 8
  ltc_xw_kernel<<<g1, 256, 0, stream>>>(x, W, b, out);

  // 2) persistent scan: 64 WGs (4 batch tiles x 16 column slices)
  ltc_scan_kernel<<<64, 256, 0, stream>>>(U, tau, out, hbuf, counters);
}